// MySAGEConvBlock_18459769438300
// MI455X (gfx1250) — compile-verified
//
#include <hip/hip_runtime.h>
#include <hip/hip_bf16.h>

typedef float v2f __attribute__((ext_vector_type(2)));
typedef float v8f __attribute__((ext_vector_type(8)));

#define DDIM 128      // feature dim
#define K2   256      // 2*D (concat)
#define AS   258      // padded LDS stride for A tile (16 rows x 256 f32)
#define HS   132      // padded LDS stride for h tile (16 rows x 128 f32)

// ---------------------------------------------------------------------------
// Pass 0: aggr = x (self loop), cnt = 1, zero BN stats
// ---------------------------------------------------------------------------
__global__ void k_init(const float* __restrict__ x, float* __restrict__ aggr,
                       float* __restrict__ cnt, float* __restrict__ stats, int N) {
    int gid = blockIdx.x * blockDim.x + threadIdx.x;
    int total4 = N * (DDIM / 4);
    if (gid < total4) ((float4*)aggr)[gid] = ((const float4*)x)[gid];
    if (gid < N) cnt[gid] = 1.0f;
    if (gid < 256) stats[gid] = 0.0f;   // colsum[128] + colsumsq[128]
}

// ---------------------------------------------------------------------------
// Pass 1: one wave32 per edge: aggr[dst] += x[src], cnt[dst] += 1
// ---------------------------------------------------------------------------
__global__ __launch_bounds__(256)
void k_edges(const int* __restrict__ src, const int* __restrict__ dst,
             const float* __restrict__ x, float* __restrict__ aggr,
             float* __restrict__ cnt, int E) {
    int tid = blockIdx.x * 256 + threadIdx.x;
    int e = tid >> 5;
    if (e >= E) return;
    int lane = tid & 31;
    int s = src[e], d = dst[e];
    float4 v = ((const float4*)x)[(size_t)s * (DDIM / 4) + lane];
    float* a = &aggr[(size_t)d * DDIM + lane * 4];
    atomicAdd(a + 0, v.x);
    atomicAdd(a + 1, v.y);
    atomicAdd(a + 2, v.z);
    atomicAdd(a + 3, v.w);
    if (lane == 0) atomicAdd(&cnt[d], 1.0f);
}

// ---------------------------------------------------------------------------
// Pass 2: per 16-row tile: h = [x | aggr/cnt] @ W + b  (f32 WMMA, K=256),
//         row-wise L2 normalize, emit h and per-column BN partial sums.
//         256 threads = 8 waves; wave w owns output columns [16w, 16w+16).
// ---------------------------------------------------------------------------
__global__ __launch_bounds__(256)
void k_gemm(const float* __restrict__ x, const float* __restrict__ aggr,
            const float* __restrict__ cnt, const float* __restrict__ W,
            const float* __restrict__ bias, float* __restrict__ hout,
            float* __restrict__ colsum, float* __restrict__ colsumsq, int N) {
    __shared__ float sA[16 * AS];
    __shared__ float sH[16 * HS];
    __shared__ float sInv[16];
    __shared__ float sRn[16];

    const int t    = threadIdx.x;
    const int lane = t & 31;
    const int wv   = t >> 5;          // wave 0..7
    const int row0 = blockIdx.x * 16;

    if (t < 16) {
        int r = row0 + t;
        sInv[t] = 1.0f / cnt[r < N ? r : (N - 1)];
    }
    __syncthreads();

    // Cooperative load of A tile: 16 rows x 256 f32 = [x_row | aggr_row * inv]
    #pragma unroll
    for (int i = 0; i < 4; ++i) {
        int idx = t + i * 256;                 // float4 index in [0,1024)
        int r   = idx >> 6;                    // 64 float4 per row
        int c4  = idx & 63;
        int gr  = row0 + r; if (gr >= N) gr = N - 1;
        float4 v;
        if (c4 < 32) {
            v = ((const float4*)x)[(size_t)gr * 32 + c4];
        } else {
            v = ((const float4*)aggr)[(size_t)gr * 32 + (c4 - 32)];
            float s = sInv[r];
            v.x *= s; v.y *= s; v.z *= s; v.w *= s;
        }
        float* p = &sA[r * AS + c4 * 4];
        p[0] = v.x; p[1] = v.y; p[2] = v.z; p[3] = v.w;
    }
    __syncthreads();

    // Fragment indices per ISA layouts (wave32):
    // A 16x4: lane&15 = M row, (lane>>4)*2 = K offset of this lane-half
    // B 4x16: lane&15 = N col, K = (lane>>4)*2 + vgpr
    const int arow = lane & 15;
    const int koff = (lane >> 4) * 2;
    const int col  = wv * 16 + (lane & 15);

    v8f acc = {};
    #pragma unroll 8
    for (int kk = 0; kk < K2 / 4; ++kk) {
        int kb = kk * 4 + koff;
        v2f a = *(const v2f*)&sA[arow * AS + kb];
        v2f b;
        b.x = W[(size_t)kb * DDIM + col];
        b.y = W[(size_t)(kb + 1) * DDIM + col];
        acc = __builtin_amdgcn_wmma_f32_16x16x4_f32(
                  false, a, false, b, (short)0, acc, false, false);
    }

    // C/D layout: VGPR v -> row v (lanes 0-15) or v+8 (lanes 16-31), col = lane&15
    float bv = bias[col];
    int rbase = (lane >> 4) * 8;
    #pragma unroll
    for (int v = 0; v < 8; ++v)
        sH[(rbase + v) * HS + col] = acc[v] + bv;
    __syncthreads();

    // Row L2 norms
    if (t < 16) {
        float ss = 0.0f;
        #pragma unroll 4
        for (int c = 0; c < DDIM; ++c) { float h = sH[t * HS + c]; ss += h * h; }
        sRn[t] = 1.0f / fmaxf(sqrtf(ss), 1e-12f);
    }
    __syncthreads();

    // Normalize, write h, accumulate BN column sums.
    // Thread t owns column c = t&127 and rows {t>>7, (t>>7)+2, ...}
    float s1 = 0.0f, s2 = 0.0f;
    int c = t & 127;
    int r0 = t >> 7;
    bool any = false;
    #pragma unroll
    for (int i = 0; i < 8; ++i) {
        int r = r0 + i * 2;
        if (row0 + r < N) {
            float hv = sH[r * HS + c] * sRn[r];
            hout[(size_t)(row0 + r) * DDIM + c] = hv;
            s1 += hv; s2 += hv * hv; any = true;
        }
    }
    if (any) {
        atomicAdd(&colsum[c], s1);
        atomicAdd(&colsumsq[c], s2);
    }
}

// ---------------------------------------------------------------------------
// Pass 3: fold BN stats into per-column scale/shift
// ---------------------------------------------------------------------------
__global__ void k_bnstats(const float* __restrict__ colsum, const float* __restrict__ colsumsq,
                          const float* __restrict__ gamma, const float* __restrict__ beta,
                          float* __restrict__ scale, float* __restrict__ shift, float invN) {
    int d = threadIdx.x;
    float mu  = colsum[d] * invN;
    float var = colsumsq[d] * invN - mu * mu;
    float sc  = gamma[d] * rsqrtf(var + 1e-5f);
    scale[d] = sc;
    shift[d] = beta[d] - mu * sc;
}

// ---------------------------------------------------------------------------
// Pass 4: out = relu(h*scale + shift + x), in place on d_out
// ---------------------------------------------------------------------------
__global__ void k_final(float* __restrict__ h, const float* __restrict__ x,
                        const float* __restrict__ scale, const float* __restrict__ shift,
                        int total4) {
    int gid = blockIdx.x * blockDim.x + threadIdx.x;
    if (gid >= total4) return;
    int c0 = (gid & (DDIM / 4 - 1)) * 4;
    float4 hv = ((float4*)h)[gid];
    float4 xv = ((const float4*)x)[gid];
    float4 o;
    o.x = fmaxf(hv.x * scale[c0 + 0] + shift[c0 + 0] + xv.x, 0.0f);
    o.y = fmaxf(hv.y * scale[c0 + 1] + shift[c0 + 1] + xv.y, 0.0f);
    o.z = fmaxf(hv.z * scale[c0 + 2] + shift[c0 + 2] + xv.z, 0.0f);
    o.w = fmaxf(hv.w * scale[c0 + 3] + shift[c0 + 3] + xv.w, 0.0f);
    ((float4*)h)[gid] = o;
}

// ---------------------------------------------------------------------------
extern "C" void kernel_launch(void* const* d_in, const int* in_sizes, int n_in,
                              void* d_out, int out_size, void* d_ws, size_t ws_size,
                              hipStream_t stream) {
    const float* x     = (const float*)d_in[0];
    const int*   ei    = (const int*)  d_in[1];
    // d_in[2] = edge_w, d_in[7] = pos_w1, d_in[8] = pos_w2: computed-but-discarded
    // in the reference (add_remaining_self_loops nulls edge_weight) -> skipped.
    const float* W     = (const float*)d_in[3];
    const float* blin  = (const float*)d_in[4];
    const float* gamma = (const float*)d_in[5];
    const float* beta  = (const float*)d_in[6];
    float* out = (float*)d_out;

    const int N = in_sizes[0] / DDIM;
    const int E = in_sizes[1] / 2;
    const int* src = ei;
    const int* dst = ei + E;

    // workspace: aggr[N*D] | cnt[N] | colsum[128] colsumsq[128] scale[128] shift[128]
    float* aggr     = (float*)d_ws;
    float* cnt      = aggr + (size_t)N * DDIM;
    float* stats    = cnt + N;
    float* colsum   = stats;
    float* colsumsq = stats + 128;
    float* scale    = stats + 256;
    float* shift    = stats + 384;

    const int total4 = N * (DDIM / 4);

    k_init<<<(total4 + 255) / 256, 256, 0, stream>>>(x, aggr, cnt, stats, N);

    long long edgeThreads = (long long)E * 32;
    k_edges<<<(unsigned)((edgeThreads + 255) / 256), 256, 0, stream>>>(
        src, dst, x, aggr, cnt, E);

    k_gemm<<<(N + 15) / 16, 256, 0, stream>>>(
        x, aggr, cnt, W, blin, out, colsum, colsumsq, N);

    k_bnstats<<<1, 128, 0, stream>>>(colsum, colsumsq, gamma, beta,
                                     scale, shift, 1.0f / (float)N);

    k_final<<<(total4 + 255) / 256, 256, 0, stream>>>(out, x, scale, shift, total4);
}